// SAGE_24584392802471
// MI455X (gfx1250) — compile-verified
//
#include <hip/hip_runtime.h>
#include <hip/hip_bf16.h>

// ---------------------------------------------------------------------------
// GraphSAGE 2-layer forward for MI455X (gfx1250, wave32).
//   Layer l: out = h @ Ws.T + mean_neigh(h) @ Wn.T + b
// Optimization: mean/linear commute -> transform first (dense WMMA GEMM),
// then aggregate the transformed features over edges (gather + fp32 atomics).
// GEMMs use V_WMMA_F32_16X16X4_F32 to keep full fp32 reference precision
// (GEMM is not the bottleneck; edge traffic ~2.5GB dominates).
// ---------------------------------------------------------------------------

typedef __attribute__((ext_vector_type(2))) float v2f;
typedef __attribute__((ext_vector_type(8))) float v8f;

__global__ void zero_kernel(float* __restrict__ p, int n) {
    int i = blockIdx.x * blockDim.x + threadIdx.x;
    if (i < n) p[i] = 0.0f;
}

__global__ void degree_kernel(const int* __restrict__ dst, float* __restrict__ deg, int E) {
    int i = blockIdx.x * blockDim.x + threadIdx.x;
    if (i < E) unsafeAtomicAdd(&deg[dst[i]], 1.0f);
}

// Per edge: gather a row of A at src, atomically accumulate into S at dst.
// cshift: log2(F/4). Consecutive threads cover consecutive float4 chunks of
// one edge -> fully coalesced gathers; fp32 hw atomics for the scatter.
__global__ void scatter_add_kernel(const float* __restrict__ A,
                                   const int* __restrict__ src,
                                   const int* __restrict__ dst,
                                   float* __restrict__ S,
                                   int total, int cshift, int F) {
    int i = blockIdx.x * blockDim.x + threadIdx.x;
    if (i >= total) return;
    int e = i >> cshift;
    int c = i & ((1 << cshift) - 1);
    int s = src[e];
    int d = dst[e];
    const float4 v = *(const float4*)(A + (size_t)s * F + c * 4);
    float* op = S + (size_t)d * F + c * 4;
    unsafeAtomicAdd(op + 0, v.x);
    unsafeAtomicAdd(op + 1, v.y);
    unsafeAtomicAdd(op + 2, v.z);
    unsafeAtomicAdd(op + 3, v.w);
}

// Y[node, 0..MT*16) = X[node, 0..128) @ W.T  (+ S/deg + bias, optional relu)
// One wave -> 16 nodes x MT*16 outputs via V_WMMA_F32_16X16X4_F32.
// W is staged in LDS in k-chunk-major layout: dword index
//   (k0/4)*ROWS*4 + col*4 + (k&3)
// so each B-fragment ds_load_b64 hits 64 distinct banks (conflict-free).
template <int MT, bool HAS_S, bool HAS_B, bool RELU>
__global__ void __launch_bounds__(256)
sage_gemm_kernel(const float* __restrict__ X, const float* __restrict__ W,
                 const float* __restrict__ bias, const float* __restrict__ S,
                 const float* __restrict__ deg, float* __restrict__ Y,
                 int ntiles) {
    constexpr int ROWS = MT * 16;           // output features
    __shared__ float sW[ROWS * 128];        // MT=8 -> 64KB, MT=4 -> 32KB

    // Cooperative swizzled load of W[ROWS][128] (row-major) into LDS.
    {
        const float4* w4 = (const float4*)W;
        float4* s4 = (float4*)sW;
        for (int i = threadIdx.x; i < ROWS * 32; i += 256) {
            int rr = i >> 5;      // feature row
            int kc = i & 31;      // k-chunk of 4
            s4[kc * ROWS + rr] = w4[i];
        }
    }
    __syncthreads();

    const int lane = threadIdx.x & 31;
    const int wid  = threadIdx.x >> 5;
    const int tile = blockIdx.x * 8 + wid;  // wave-uniform -> EXEC all-1s
    if (tile >= ntiles) return;

    const int r  = lane & 15;        // A: node row in tile / B: feature col
    const int kh = (lane >> 4) * 2;  // k sub-offset (0 or 2)

    const float* xrow = X + (size_t)(tile * 16 + r) * 128 + kh;

    v8f acc[MT] = {};
#pragma unroll
    for (int k0 = 0; k0 < 128; k0 += 4) {
        v2f a;
        a.x = xrow[k0];
        a.y = xrow[k0 + 1];
        const float* wk = sW + (k0 >> 2) * (ROWS * 4) + kh;
#pragma unroll
        for (int mt = 0; mt < MT; ++mt) {
            const float* wp = wk + (mt * 16 + r) * 4;
            v2f b;
            b.x = wp[0];
            b.y = wp[1];
            acc[mt] = __builtin_amdgcn_wmma_f32_16x16x4_f32(
                false, a, false, b, (short)0, acc[mt], false, false);
        }
    }

    // Epilogue. C/D layout: VGPR j = node (j + 8*(lane>>4)), lane&15 = feature.
    const int M  = ROWS;
    const int nb = (lane >> 4) * 8;
    float rdeg[8];
    if (HAS_S) {
#pragma unroll
        for (int j = 0; j < 8; ++j) {
            float dg = deg[tile * 16 + nb + j];
            rdeg[j] = 1.0f / fmaxf(dg, 1.0f);
        }
    }
#pragma unroll
    for (int mt = 0; mt < MT; ++mt) {
        const int f = mt * 16 + r;
        const float bv = HAS_B ? bias[f] : 0.0f;
#pragma unroll
        for (int j = 0; j < 8; ++j) {
            const size_t node = (size_t)(tile * 16 + nb + j);
            float v = acc[mt][j] + bv;
            if (HAS_S) v += S[node * M + f] * rdeg[j];
            if (RELU) v = fmaxf(v, 0.0f);
            Y[node * M + f] = v;
        }
    }
}

extern "C" void kernel_launch(void* const* d_in, const int* in_sizes, int n_in,
                              void* d_out, int out_size, void* d_ws, size_t ws_size,
                              hipStream_t stream) {
    const float* x   = (const float*)d_in[0];
    const int*   src = (const int*)d_in[1];
    const int*   dst = (const int*)d_in[2];
    const float* W1s = (const float*)d_in[3];
    const float* W1n = (const float*)d_in[4];
    const float* b1  = (const float*)d_in[5];
    const float* W2s = (const float*)d_in[6];
    const float* W2n = (const float*)d_in[7];
    const float* b2  = (const float*)d_in[8];
    float* out = (float*)d_out;

    const int N = in_sizes[0] / 128;  // 100000 (divisible by 16)
    const int E = in_sizes[1];        // 1600000

    // Workspace layout (floats): deg[N] | bufA[N*128] | bufB[N*128]
    // bufA: a1 (layer1 neigh-transform), then reused for h.
    // bufB: s1 (layer1 scatter sum), then reused as a2 (low) + s2 (high).
    float* ws   = (float*)d_ws;
    float* deg  = ws;
    float* bufA = ws + N;
    float* bufB = bufA + (size_t)N * 128;
    float* s1 = bufB;
    float* a2 = bufB;
    float* s2 = bufB + (size_t)N * 64;

    const dim3 blk(256);
    const int ntiles = (N + 15) / 16;
    const int gblocks = (ntiles + 7) / 8;

    // ---- Layer 1 ----
    zero_kernel<<<(N + 255) / 256, blk, 0, stream>>>(deg, N);
    zero_kernel<<<((N * 128) + 255) / 256, blk, 0, stream>>>(s1, N * 128);
    degree_kernel<<<(E + 255) / 256, blk, 0, stream>>>(dst, deg, E);

    // a1 = x @ W1n.T
    sage_gemm_kernel<8, false, false, false><<<gblocks, blk, 0, stream>>>(
        x, W1n, nullptr, nullptr, nullptr, bufA, ntiles);
    // s1[dst] += a1[src]
    {
        int total = E * 32;  // 128/4 chunks per edge
        scatter_add_kernel<<<(total + 255) / 256, blk, 0, stream>>>(
            bufA, src, dst, s1, total, 5, 128);
    }
    // h = relu(x @ W1s.T + s1/deg + b1)   (overwrites bufA)
    sage_gemm_kernel<8, true, true, true><<<gblocks, blk, 0, stream>>>(
        x, W1s, b1, s1, deg, bufA, ntiles);

    // ---- Layer 2 ----
    zero_kernel<<<((N * 64) + 255) / 256, blk, 0, stream>>>(s2, N * 64);
    // a2 = h @ W2n.T
    sage_gemm_kernel<4, false, false, false><<<gblocks, blk, 0, stream>>>(
        bufA, W2n, nullptr, nullptr, nullptr, a2, ntiles);
    // s2[dst] += a2[src]
    {
        int total = E * 16;  // 64/4 chunks per edge
        scatter_add_kernel<<<(total + 255) / 256, blk, 0, stream>>>(
            a2, src, dst, s2, total, 4, 64);
    }
    // out = h @ W2s.T + s2/deg + b2
    sage_gemm_kernel<4, true, true, false><<<gblocks, blk, 0, stream>>>(
        bufA, W2s, b2, s2, deg, out, ntiles);
}